// PoseTokenizer_56057913147981
// MI455X (gfx1250) — compile-verified
//
#include <hip/hip_runtime.h>
#include <hip/hip_bf16.h>

typedef float v2f __attribute__((ext_vector_type(2)));
typedef float v8f __attribute__((ext_vector_type(8)));

#define N_PARTS   24
#define PART_DIM  6
#define TOKEN_DIM 256
#define BATCH     32768

#define XROW      (N_PARTS * PART_DIM)        // 144 floats per x row
#define OROW      (N_PARTS * TOKEN_DIM)       // 6144 floats per out row
#define CT_PER_PART (TOKEN_DIM / 16)          // 16 column tiles per part
#define R_ITERS   16                          // row tiles per wave
#define ROW_GROUPS (BATCH / 16 / R_ITERS)     // 128
#define WAVES_PER_BLOCK 8

// Transposed-tile mapping: D[M=t, N=b] = sum_k W[n,k,t] * x[b, n*6+k] + bias[n,t]
//   A = 16x4 slice of W^T (loop-invariant), B = 4x16 slice of x^T,
//   C = bias (loop-invariant, varies along VGPR index -> one v8f load).
// C/D layout => each lane's 8 accumulators are 8 CONSECUTIVE t values of one
// batch row: the tile stores as two global_store_b128 per lane-group.
// K=6 padded to 8 by zeroing A's K={6,7} lanes once (B garbage * 0 == 0).
__global__ __launch_bounds__(256) void pose_tokenizer_wmma_f32(
    const float* __restrict__ x,     // [BATCH, 144]
    const float* __restrict__ W,     // [24, 6, 256]
    const float* __restrict__ bias,  // [24, 256]
    float* __restrict__ out)         // [BATCH, 24, 256]
{
    const int lane = threadIdx.x & 31;
    const int wave = threadIdx.x >> 5;
    const int tile = blockIdx.x * WAVES_PER_BLOCK + wave;

    const int ct = tile & (CT_PER_PART - 1);                 // column tile
    const int n  = (tile / CT_PER_PART) % N_PARTS;           // part
    const int rg = tile / (CT_PER_PART * N_PARTS);           // row group

    const int m0   = rg * (16 * R_ITERS);                    // first batch row
    const int t0   = ct * 16;                                // first token col
    const int half = lane >> 4;     // 0: lanes 0-15 (K lo), 1: lanes 16-31 (K hi)
    const int l15  = lane & 15;

    // ---- A = W^T tile (16x4 f32), loop-invariant ----
    // lane -> M = t0+l15; VGPR0/1 = K pair selected by half.
    const float* wcol = W + (size_t)n * PART_DIM * TOKEN_DIM + (t0 + l15);
    v2f a1, a2;
    {
        const int kb = 2 * half;                 // K = {0,1} | {2,3}
        a1.x = wcol[(size_t)(kb + 0) * TOKEN_DIM];
        a1.y = wcol[(size_t)(kb + 1) * TOKEN_DIM];
    }
    a2.x = wcol[(size_t)4 * TOKEN_DIM];          // K = 4
    a2.y = wcol[(size_t)5 * TOKEN_DIM];          // K = 5
    if (half) { a2.x = 0.0f; a2.y = 0.0f; }      // K = {6,7}: zero pad (once)

    // ---- C = bias, loop-invariant: cbias[i] = bias[n, t0 + 8*half + i] ----
    const v8f cbias = *(const v8f*)(bias + (size_t)n * TOKEN_DIM + t0 + 8 * half);

    // ---- bases (all loop offsets become 24-bit immediates) ----
    // B tile: lane -> N = batch row m0+l15; VGPR0/1 = K pair by half.
    const float* xbase = x + (size_t)(m0 + l15) * XROW + (size_t)n * PART_DIM;
    float* obase = out + (size_t)(m0 + l15) * OROW
                       + (size_t)n * TOKEN_DIM + t0 + 8 * half;

    #pragma unroll
    for (int i = 0; i < R_ITERS; ++i) {
        const float* xr = xbase + (size_t)i * (16 * XROW);
        // K = {0,1} (half 0) or {2,3} (half 1)
        v2f b1 = *(const v2f*)(xr + 2 * half);
        // K = {4,5}; half-1 lanes load {4,5} too (in-bounds; A side is zero)
        v2f b2 = *(const v2f*)(xr + 4);

        v8f d = __builtin_amdgcn_wmma_f32_16x16x4_f32(false, a2, false, b2,
                                                      (short)0, cbias, false, false);
        d     = __builtin_amdgcn_wmma_f32_16x16x4_f32(false, a1, false, b1,
                                                      (short)0, d, false, false);

        // 8 consecutive t values per lane -> one 32B store (2x b128)
        *(v8f*)(obase + (size_t)i * (16 * OROW)) = d;
    }
}

extern "C" void kernel_launch(void* const* d_in, const int* in_sizes, int n_in,
                              void* d_out, int out_size, void* d_ws, size_t ws_size,
                              hipStream_t stream) {
    const float* x  = (const float*)d_in[0];  // [32768, 144]
    const float* W  = (const float*)d_in[1];  // [24, 6, 256]
    const float* b  = (const float*)d_in[2];  // [24, 256]
    float* out = (float*)d_out;               // [32768, 24, 256]

    const int total_waves = ROW_GROUPS * N_PARTS * CT_PER_PART;   // 49152
    const int blocks = total_waves / WAVES_PER_BLOCK;             // 6144
    pose_tokenizer_wmma_f32<<<blocks, 256, 0, stream>>>(x, W, b, out);
}